// GCNClassifier_4174708212139
// MI455X (gfx1250) — compile-verified
//
#include <hip/hip_runtime.h>

typedef __attribute__((ext_vector_type(2))) float v2f;
typedef __attribute__((ext_vector_type(8))) float v8f;

#define NNODES 100000
#define D_IN   512
#define H1F    128
#define H2F    16
#define NCLS   10
#define NSTRIP (NNODES / 16)   // 6250 row-strips of 16

// ---------------------------------------------------------------- degree/norm
__global__ void k_init_deg(float* deg) {
    int i = blockIdx.x * blockDim.x + threadIdx.x;
    if (i < NNODES) deg[i] = 1.0f;                  // self-loop contribution
}

__global__ void k_deg_edges(const long long* __restrict__ ei, int E, float* deg) {
    int e = blockIdx.x * blockDim.x + threadIdx.x;
    if (e < E) {
        long long d = ei[(size_t)E + e];
        unsafeAtomicAdd(&deg[d], 1.0f);             // global_atomic_add_f32
    }
}

__global__ void k_dinv(float* deg) {
    int i = blockIdx.x * blockDim.x + threadIdx.x;
    if (i < NNODES) {
        float d = deg[i];
        deg[i] = (d > 0.0f) ? rsqrtf(d) : 0.0f;     // in place: deg -> dinv
    }
}

// ---------------------------------------------------------------- GEMM1: X@W1
// Wave-per-16-row-strip, all 128 output cols (8 accumulators of 16x16).
// W1 staged in LDS in 64KB k-chunks, packed as (k,k+1)-pairs per n so B
// fragments are single 8-byte LDS loads.
__global__ void __launch_bounds__(256)
k_gemm1(const float* __restrict__ X, const float* __restrict__ W,
        float* __restrict__ H) {
    __shared__ float ldsW[64 * 256];                // 64 k-pairs x 128 n x 2 = 64KB
    const int lane = threadIdx.x & 31;
    const int wave = threadIdx.x >> 5;
    const int strip = blockIdx.x * 8 + wave;
    const bool active = strip < NSTRIP;
    const int l16 = lane & 15;
    const int hi  = lane >> 4;                      // lane half: 0 or 1
    const size_t m0 = (size_t)strip * 16;
    const float* xrow = active ? (X + (m0 + l16) * D_IN) : X;

    v8f acc[8];
#pragma unroll
    for (int t = 0; t < 8; ++t) acc[t] = (v8f){};

    for (int kc = 0; kc < D_IN; kc += 128) {
        __syncthreads();
        // cooperative, coalesced stage of W1[kc:kc+128][0:128] as k-pairs
        for (int idx = threadIdx.x; idx < 64 * 128; idx += 256) {
            int n  = idx & 127;
            int kp = idx >> 7;
            int kk = kc + kp * 2;
            ldsW[kp * 256 + n * 2 + 0] = W[(size_t)kk * H1F + n];
            ldsW[kp * 256 + n * 2 + 1] = W[(size_t)(kk + 1) * H1F + n];
        }
        __syncthreads();
        if (active) {
#pragma unroll 4
            for (int k = 0; k < 128; k += 4) {
                v2f a = *(const v2f*)(xrow + kc + k + 2 * hi);
                const float* bb = &ldsW[((k >> 1) + hi) * 256];
#pragma unroll
                for (int t = 0; t < 8; ++t) {
                    v2f b = *(const v2f*)(bb + (t * 16 + l16) * 2);
                    acc[t] = __builtin_amdgcn_wmma_f32_16x16x4_f32(
                        false, a, false, b, (short)0, acc[t], false, false);
                }
            }
        }
    }
    if (active) {
#pragma unroll
        for (int t = 0; t < 8; ++t)
#pragma unroll
            for (int r = 0; r < 8; ++r) {
                size_t row = m0 + r + 8 * hi;
                H[row * H1F + t * 16 + l16] = acc[t][r];
            }
    }
}

// ---------------------------------------------------------------- GEMM2: feat1@W2
__global__ void __launch_bounds__(256)
k_gemm2(const float* __restrict__ F1, const float* __restrict__ W,
        float* __restrict__ H) {
    __shared__ float ldsW[64 * 32];                 // 64 k-pairs x 16 n x 2 = 8KB
    const int lane = threadIdx.x & 31;
    const int wave = threadIdx.x >> 5;
    const int strip = blockIdx.x * 8 + wave;
    const bool active = strip < NSTRIP;
    const int l16 = lane & 15;
    const int hi  = lane >> 4;
    const size_t m0 = (size_t)strip * 16;

    for (int idx = threadIdx.x; idx < 64 * 16; idx += 256) {
        int n  = idx & 15;
        int kp = idx >> 4;
        ldsW[kp * 32 + n * 2 + 0] = W[(size_t)(2 * kp) * H2F + n];
        ldsW[kp * 32 + n * 2 + 1] = W[(size_t)(2 * kp + 1) * H2F + n];
    }
    __syncthreads();
    if (!active) return;

    const float* frow = F1 + (m0 + l16) * H1F;
    v8f acc = (v8f){};
#pragma unroll 4
    for (int k = 0; k < H1F; k += 4) {
        v2f a = *(const v2f*)(frow + k + 2 * hi);
        v2f b = *(const v2f*)(&ldsW[((k >> 1) + hi) * 32 + l16 * 2]);
        acc = __builtin_amdgcn_wmma_f32_16x16x4_f32(
            false, a, false, b, (short)0, acc, false, false);
    }
#pragma unroll
    for (int r = 0; r < 8; ++r) {
        size_t row = m0 + r + 8 * hi;
        H[row * H2F + l16] = acc[r];
    }
}

// ---------------------------------------------------------------- aggregation
template <int F>
__global__ void k_agg_init(const float* __restrict__ dinv,
                           const float* __restrict__ Ht, float* __restrict__ A) {
    size_t i = (size_t)blockIdx.x * blockDim.x + threadIdx.x;
    if (i < (size_t)NNODES * F) {
        float di = dinv[i / F];
        A[i] = di * di * Ht[i];                     // self-loop term, norm = dinv^2
    }
}

// one wave per edge, 128 features = float4 per lane
__global__ void k_agg_edges128(const long long* __restrict__ ei, int E,
                               const float* __restrict__ dinv,
                               const float* __restrict__ Ht, float* __restrict__ A) {
    int wid  = (blockIdx.x * blockDim.x + threadIdx.x) >> 5;
    int lane = threadIdx.x & 31;
    if (wid >= E) return;
    long long s = ei[wid];
    long long d = ei[(size_t)E + wid];
    float nrm = dinv[s] * dinv[d];
    float4 v = ((const float4*)(Ht + (size_t)s * H1F))[lane];
    float* out = A + (size_t)d * H1F + lane * 4;
    unsafeAtomicAdd(out + 0, nrm * v.x);
    unsafeAtomicAdd(out + 1, nrm * v.y);
    unsafeAtomicAdd(out + 2, nrm * v.z);
    unsafeAtomicAdd(out + 3, nrm * v.w);
}

// thread per (edge, feature), 16 features
__global__ void k_agg_edges16(const long long* __restrict__ ei, int E,
                              const float* __restrict__ dinv,
                              const float* __restrict__ Ht, float* __restrict__ A) {
    size_t tid = (size_t)blockIdx.x * blockDim.x + threadIdx.x;
    if (tid >= (size_t)E * H2F) return;
    int e = (int)(tid >> 4);
    int f = (int)(tid & 15);
    long long s = ei[e];
    long long d = ei[(size_t)E + e];
    float nrm = dinv[s] * dinv[d];
    unsafeAtomicAdd(&A[(size_t)d * H2F + f], nrm * Ht[(size_t)s * H2F + f]);
}

// ---------------------------------------------------------------- bias+PReLU+dropout
template <int F>
__global__ void k_ew(const float* __restrict__ A, const float* __restrict__ bias,
                     const float* __restrict__ aScalar, const float* __restrict__ u,
                     float* __restrict__ out) {
    size_t i = (size_t)blockIdx.x * blockDim.x + threadIdx.x;
    if (i < (size_t)NNODES * F) {
        float h = A[i] + bias[i & (F - 1)];
        h = (h > 0.0f) ? h : aScalar[0] * h;
        out[i] = (u[i] >= 0.5f) ? h * 2.0f : 0.0f;  // dropout p=0.5
    }
}

// ---------------------------------------------------------------- classifier head
__global__ void k_logits(const float* __restrict__ feat2, const float* __restrict__ fcW,
                         const float* __restrict__ fcb, float* __restrict__ logits) {
    size_t tid = (size_t)blockIdx.x * blockDim.x + threadIdx.x;
    if (tid >= (size_t)NNODES * NCLS) return;
    size_t n = tid / NCLS;
    int    c = (int)(tid % NCLS);
    const float* f = feat2 + n * H2F;
    float s = fcb[c];
#pragma unroll
    for (int k = 0; k < H2F; ++k) s += f[k] * fcW[k * NCLS + c];
    logits[tid] = s;
}

// ---------------------------------------------------------------- launch
extern "C" void kernel_launch(void* const* d_in, const int* in_sizes, int n_in,
                              void* d_out, int out_size, void* d_ws, size_t ws_size,
                              hipStream_t stream) {
    const float*     x   = (const float*)d_in[0];
    const long long* ei  = (const long long*)d_in[1];
    const float*     W1  = (const float*)d_in[2];
    const float*     b1  = (const float*)d_in[3];
    const float*     W2  = (const float*)d_in[4];
    const float*     b2  = (const float*)d_in[5];
    const float*     aP  = (const float*)d_in[6];
    const float*     fcW = (const float*)d_in[7];
    const float*     fcb = (const float*)d_in[8];
    const float*     u1  = (const float*)d_in[9];
    const float*     u2  = (const float*)d_in[10];
    const int E = in_sizes[1] / 2;

    char* ws = (char*)d_ws;
    float* dinv = (float*)(ws + 0);
    float* Ht1  = (float*)(ws + (1u << 20));                     // 51.2 MB
    float* A1   = (float*)(ws + (1u << 20) + 52428800u);         // 51.2 MB (becomes feat1)
    float* Ht2  = (float*)(ws + (1u << 20));                     // reuses dead Ht1 slot
    float* A2   = (float*)(ws + (9u << 20));                     // 6.4 MB
    float* feat2  = (float*)d_out;
    float* logits = (float*)d_out + (size_t)NNODES * H2F;

    const int T = 256;
    // normalization
    k_init_deg<<<(NNODES + T - 1) / T, T, 0, stream>>>(dinv);
    k_deg_edges<<<(E + T - 1) / T, T, 0, stream>>>(ei, E, dinv);
    k_dinv<<<(NNODES + T - 1) / T, T, 0, stream>>>(dinv);
    // layer 1
    k_gemm1<<<(NSTRIP + 7) / 8, T, 0, stream>>>(x, W1, Ht1);
    k_agg_init<H1F><<<((size_t)NNODES * H1F + T - 1) / T, T, 0, stream>>>(dinv, Ht1, A1);
    k_agg_edges128<<<((size_t)E * 32 + T - 1) / T, T, 0, stream>>>(ei, E, dinv, Ht1, A1);
    k_ew<H1F><<<((size_t)NNODES * H1F + T - 1) / T, T, 0, stream>>>(A1, b1, aP, u1, A1);
    // layer 2
    k_gemm2<<<(NSTRIP + 7) / 8, T, 0, stream>>>(A1, W2, Ht2);
    k_agg_init<H2F><<<((size_t)NNODES * H2F + T - 1) / T, T, 0, stream>>>(dinv, Ht2, A2);
    k_agg_edges16<<<((size_t)E * H2F + T - 1) / T, T, 0, stream>>>(ei, E, dinv, Ht2, A2);
    k_ew<H2F><<<((size_t)NNODES * H2F + T - 1) / T, T, 0, stream>>>(A2, b2, aP, u2, feat2);
    // head
    k_logits<<<((size_t)NNODES * NCLS + T - 1) / T, T, 0, stream>>>(feat2, fcW, fcb, logits);
}